// SplineAttention_65034394796221
// MI455X (gfx1250) — compile-verified
//
#include <hip/hip_runtime.h>

typedef __attribute__((ext_vector_type(2))) float v2f;
typedef __attribute__((ext_vector_type(8))) float v8f;
typedef int v4i_vs __attribute__((vector_size(16)));   // matches builtin param type

#define S_LEN 2048
#define DMODEL 1024
#define HEADS 16
#define HDIM 64

#if __has_builtin(__builtin_amdgcn_global_load_async_to_lds_b128) && \
    __has_builtin(__builtin_amdgcn_s_wait_asynccnt)
#define USE_ASYNC_LDS 1
#else
#define USE_ASYNC_LDS 0
#endif

__device__ __forceinline__ v8f wmma_f32(v2f a, v2f b, v8f c) {
  // V_WMMA_F32_16X16X4_F32 : D = A(16x4) * B(4x16) + C(16x16), all fp32
  return __builtin_amdgcn_wmma_f32_16x16x4_f32(
      /*neg_a=*/false, a, /*neg_b=*/false, b,
      /*c_mod=*/(short)0, c, /*reuse_a=*/false, /*reuse_b=*/false);
}

// 16-byte global -> LDS transfer. Async (no VGPR bounce, ASYNCcnt-tracked) when
// the toolchain exposes the gfx1250 builtin; plain float4 copy otherwise.
__device__ __forceinline__ void cp16(const float* g, float* l) {
#if USE_ASYNC_LDS
  __builtin_amdgcn_global_load_async_to_lds_b128(
      (__attribute__((address_space(1))) v4i_vs*)(uintptr_t)g,
      (__attribute__((address_space(3))) v4i_vs*)l, 0, 0);
#else
  *(float4*)l = *(const float4*)g;
#endif
}
__device__ __forceinline__ void wait_async_keep4() {
#if USE_ASYNC_LDS
  __builtin_amdgcn_s_wait_asynccnt(4);   // all but the newest 4 ops complete
#endif
}
__device__ __forceinline__ void wait_async_all() {
#if USE_ASYNC_LDS
  __builtin_amdgcn_s_wait_asynccnt(0);
#endif
}

// ---------------------------------------------------------------------------
// GEMM: C[M][N] = X[M][K] @ W[N][K]^T + bias[N]      (torch Linear semantics)
// split_heads=1 scatters C into [N/64][M][64] so attention reads per-head rows.
// 128 threads = 4 waves; block tile 64x64; K staged 16-wide, double-buffered
// LDS with async global->LDS copies overlapping the WMMA chain.
// ---------------------------------------------------------------------------
__global__ __launch_bounds__(128)
void gemm_bias_wmma(const float* __restrict__ X, const float* __restrict__ W,
                    const float* __restrict__ bias, float* __restrict__ C,
                    int M, int N, int K, int split_heads) {
  __shared__ float Xs[2][64][20];   // pitch 20: 16B-aligned rows, bank-safe
  __shared__ float Ws[2][64][20];

  const int tid  = threadIdx.x;
  const int w    = tid >> 5;
  const int lane = tid & 31;
  const int half = lane >> 4;
  const int lr   = lane & 15;
  const int m0 = blockIdx.y * 64;
  const int n0 = blockIdx.x * 64;

  // staging coords: this thread moves rows r0 and r0+32, 16B column c0 (x2 mats)
  const int r0 = tid >> 2;
  const int c0 = (tid & 3) * 4;
  const float* xg0 = X + (size_t)(m0 + r0) * K + c0;
  const float* xg1 = X + (size_t)(m0 + r0 + 32) * K + c0;
  const float* wg0 = W + (size_t)(n0 + r0) * K + c0;
  const float* wg1 = W + (size_t)(n0 + r0 + 32) * K + c0;

  v8f acc0 = {0,0,0,0,0,0,0,0};
  v8f acc1 = acc0, acc2 = acc0, acc3 = acc0;

  const int T = K / 16;
  // prologue: issue tile 0 into buffer 0
  cp16(xg0, &Xs[0][r0][c0]);
  cp16(xg1, &Xs[0][r0 + 32][c0]);
  cp16(wg0, &Ws[0][r0][c0]);
  cp16(wg1, &Ws[0][r0 + 32][c0]);

  for (int i = 0; i < T; ++i) {
    const int cur = i & 1;
    __syncthreads();                       // everyone done reading buf cur^1
    if (i + 1 < T) {                       // stream tile i+1 while computing i
      const int nxt = cur ^ 1;
      const int ko = (i + 1) * 16;
      cp16(xg0 + ko, &Xs[nxt][r0][c0]);
      cp16(xg1 + ko, &Xs[nxt][r0 + 32][c0]);
      cp16(wg0 + ko, &Ws[nxt][r0][c0]);
      cp16(wg1 + ko, &Ws[nxt][r0 + 32][c0]);
      wait_async_keep4();                  // tile i landed (in-order completion)
    } else {
      wait_async_all();
    }
    __syncthreads();                       // tile i visible block-wide

#pragma unroll
    for (int c = 0; c < 4; ++c) {
      // A layout (16x4 f32): v0 = {K=4c+0 | lanes<16, K=4c+2 | lanes>=16}, v1 = +1
      v2f a;
      a.x = Xs[cur][16 * w + lr][4 * c + 2 * half];
      a.y = Xs[cur][16 * w + lr][4 * c + 2 * half + 1];
      v2f b;
      b.x = Ws[cur][ 0 + lr][4 * c + 2 * half]; b.y = Ws[cur][ 0 + lr][4 * c + 2 * half + 1];
      acc0 = wmma_f32(a, b, acc0);
      b.x = Ws[cur][16 + lr][4 * c + 2 * half]; b.y = Ws[cur][16 + lr][4 * c + 2 * half + 1];
      acc1 = wmma_f32(a, b, acc1);
      b.x = Ws[cur][32 + lr][4 * c + 2 * half]; b.y = Ws[cur][32 + lr][4 * c + 2 * half + 1];
      acc2 = wmma_f32(a, b, acc2);
      b.x = Ws[cur][48 + lr][4 * c + 2 * half]; b.y = Ws[cur][48 + lr][4 * c + 2 * half + 1];
      acc3 = wmma_f32(a, b, acc3);
    }
  }

#pragma unroll
  for (int r = 0; r < 8; ++r) {
    const int m = m0 + 16 * w + 8 * half + r;   // C layout: vgpr r -> row r (+8 upper half)
    float vals[4] = {acc0[r], acc1[r], acc2[r], acc3[r]};
#pragma unroll
    for (int t = 0; t < 4; ++t) {
      const int n = n0 + 16 * t + lr;
      const float o = vals[t] + bias[n];
      if (split_heads) {
        C[((size_t)(n >> 6) * M + m) * HDIM + (n & 63)] = o;
      } else {
        C[(size_t)m * N + n] = o;
      }
    }
  }
}

// ---------------------------------------------------------------------------
// Flash-style spline attention. Q/K/V in [H][S][64] layout.
// One block (4 waves / 128 thr) per (head, 64-query tile); wave w owns 16 rows.
// K/V tiles double-buffered with async global->LDS streaming.
// ---------------------------------------------------------------------------
__global__ __launch_bounds__(128)
void spline_flash_attn(const float* __restrict__ Qg, const float* __restrict__ Kg,
                       const float* __restrict__ Vg, const float* __restrict__ cf_g,
                       float* __restrict__ Og) {
  __shared__ float Qs[64][68];
  __shared__ float Ks[2][16][68];
  __shared__ float Vs[2][16][68];
  __shared__ float Ps[4][16][20];   // per-wave exp(S) tile, C-layout -> A-layout bounce

  const int tid  = threadIdx.x;
  const int w    = tid >> 5;
  const int lane = tid & 31;
  const int half = lane >> 4;
  const int lr   = lane & 15;
  const int h  = blockIdx.y;
  const int q0 = blockIdx.x * 64;

  // spline intervals: t=linspace(0,1,10); lo=t[max(0,i-2)], hi=t[min(i+1,9)]
  const float LO[12] = {0.f, 0.f, 0.f, 1.f/9.f, 2.f/9.f, 3.f/9.f, 4.f/9.f,
                        5.f/9.f, 6.f/9.f, 7.f/9.f, 8.f/9.f, 1.f};
  const float HI[12] = {1.f/9.f, 2.f/9.f, 3.f/9.f, 4.f/9.f, 5.f/9.f, 6.f/9.f,
                        7.f/9.f, 8.f/9.f, 1.f, 1.f, 1.f, 1.f};
  float cf[12];
#pragma unroll
  for (int i = 0; i < 12; ++i) cf[i] = cf_g[i];

  // staging coords: rows kr0 / kr0+8, 16B column kc0
  const int kr0 = tid >> 4;
  const int kc0 = (tid & 15) * 4;

  // stage Q tile [64][64]: 8 transfers/thread (rows kr0 + 8*i)
  const float* Qbase = Qg + ((size_t)h * S_LEN + q0) * HDIM;
#pragma unroll
  for (int i = 0; i < 8; ++i)
    cp16(Qbase + (size_t)(kr0 + 8 * i) * HDIM + kc0, &Qs[kr0 + 8 * i][kc0]);

  const float* kg0 = Kg + ((size_t)h * S_LEN + kr0) * HDIM + kc0;
  const float* kg1 = kg0 + 8 * HDIM;
  const float* vg0 = Vg + ((size_t)h * S_LEN + kr0) * HDIM + kc0;
  const float* vg1 = vg0 + 8 * HDIM;

  // prologue: issue K/V tile 0 into buffer 0
  cp16(kg0, &Ks[0][kr0][kc0]);
  cp16(kg1, &Ks[0][kr0 + 8][kc0]);
  cp16(vg0, &Vs[0][kr0][kc0]);
  cp16(vg1, &Vs[0][kr0 + 8][kc0]);

  v8f o0 = {0,0,0,0,0,0,0,0};
  v8f o1 = o0, o2 = o0, o3 = o0;
  float m_run[8], l_run[8];
#pragma unroll
  for (int r = 0; r < 8; ++r) { m_run[r] = -1e30f; l_run[r] = 0.f; }

  const int T = S_LEN / 16;                // 128 key tiles
  for (int j = 0; j < T; ++j) {
    const int cur = j & 1;
    __syncthreads();                       // done reading buf cur^1
    if (j + 1 < T) {
      const int nxt = cur ^ 1;
      const size_t jo = (size_t)(j + 1) * 16 * HDIM;
      cp16(kg0 + jo, &Ks[nxt][kr0][kc0]);
      cp16(kg1 + jo, &Ks[nxt][kr0 + 8][kc0]);
      cp16(vg0 + jo, &Vs[nxt][kr0][kc0]);
      cp16(vg1 + jo, &Vs[nxt][kr0 + 8][kc0]);
      wait_async_keep4();                  // tile j (and Q) landed, in-order
    } else {
      wait_async_all();
    }
    __syncthreads();

    // S = Qw(16x64) @ Kj^T(64x16) : 16 chained f32 WMMAs over K-dim
    v8f s = {0,0,0,0,0,0,0,0};
#pragma unroll
    for (int c = 0; c < 16; ++c) {
      v2f a, b;
      a.x = Qs[16 * w + lr][4 * c + 2 * half];
      a.y = Qs[16 * w + lr][4 * c + 2 * half + 1];
      b.x = Ks[cur][lr][4 * c + 2 * half];      // B[k][n] = K[n][k]
      b.y = Ks[cur][lr][4 * c + 2 * half + 1];
      s = wmma_f32(a, b, s);
    }

    // scale + spline (branchless bin lookup)
    float pv[8];
#pragma unroll
    for (int r = 0; r < 8; ++r) {
      float sv = s[r] * 0.125f;                 // 1/sqrt(64)
      float y = 0.f;
#pragma unroll
      for (int i = 0; i < 12; ++i)
        y += ((sv >= LO[i]) && (sv < HI[i])) ? cf[i] : 0.f;
      pv[r] = y;
    }

    // online softmax per row (rows split: lanes<16 -> r, lanes>=16 -> r+8)
#pragma unroll
    for (int r = 0; r < 8; ++r) {
      float rmax = pv[r];
#pragma unroll
      for (int off = 8; off; off >>= 1)
        rmax = fmaxf(rmax, __shfl_xor(rmax, off, 32));
      const float m_new = fmaxf(m_run[r], rmax);
      const float corr  = __expf(m_run[r] - m_new);
      const float p     = __expf(pv[r] - m_new);
      float rsum = p;
#pragma unroll
      for (int off = 8; off; off >>= 1)
        rsum += __shfl_xor(rsum, off, 32);
      l_run[r] = l_run[r] * corr + rsum;
      m_run[r] = m_new;
      o0[r] *= corr; o1[r] *= corr; o2[r] *= corr; o3[r] *= corr;
      Ps[w][8 * half + r][lr] = p;              // C-layout -> LDS row-major
    }
    __syncthreads();                            // fence Ps stores vs A-layout reads

    // O += P(16x16) @ Vj(16x64)
#pragma unroll
    for (int c = 0; c < 4; ++c) {
      v2f a, b;
      a.x = Ps[w][lr][4 * c + 2 * half];
      a.y = Ps[w][lr][4 * c + 2 * half + 1];
      b.x = Vs[cur][4 * c + 2 * half][ 0 + lr]; b.y = Vs[cur][4 * c + 2 * half + 1][ 0 + lr];
      o0 = wmma_f32(a, b, o0);
      b.x = Vs[cur][4 * c + 2 * half][16 + lr]; b.y = Vs[cur][4 * c + 2 * half + 1][16 + lr];
      o1 = wmma_f32(a, b, o1);
      b.x = Vs[cur][4 * c + 2 * half][32 + lr]; b.y = Vs[cur][4 * c + 2 * half + 1][32 + lr];
      o2 = wmma_f32(a, b, o2);
      b.x = Vs[cur][4 * c + 2 * half][48 + lr]; b.y = Vs[cur][4 * c + 2 * half + 1][48 + lr];
      o3 = wmma_f32(a, b, o3);
    }
  }

  // normalize and write out as [S][D] row-major (head h -> cols h*64..)
#pragma unroll
  for (int r = 0; r < 8; ++r) {
    const int m = q0 + 16 * w + 8 * half + r;
    const float inv = 1.f / l_run[r];
    const size_t base = (size_t)m * DMODEL + (size_t)h * HDIM;
    Og[base +  0 + lr] = o0[r] * inv;
    Og[base + 16 + lr] = o1[r] * inv;
    Og[base + 32 + lr] = o2[r] * inv;
    Og[base + 48 + lr] = o3[r] * inv;
  }
}

// ---------------------------------------------------------------------------
extern "C" void kernel_launch(void* const* d_in, const int* in_sizes, int n_in,
                              void* d_out, int out_size, void* d_ws, size_t ws_size,
                              hipStream_t stream) {
  (void)in_sizes; (void)n_in; (void)out_size; (void)ws_size;
  const float* x  = (const float*)d_in[0];
  const float* Wq = (const float*)d_in[1];
  const float* bq = (const float*)d_in[2];
  const float* Wk = (const float*)d_in[3];
  const float* bk = (const float*)d_in[4];
  const float* Wv = (const float*)d_in[5];
  const float* bv = (const float*)d_in[6];
  const float* Wo = (const float*)d_in[7];
  const float* bo = (const float*)d_in[8];
  const float* cf = (const float*)d_in[9];

  float* Qb = (float*)d_ws;                       // [16][2048][64]
  float* Kb = Qb + (size_t)S_LEN * DMODEL;
  float* Vb = Kb + (size_t)S_LEN * DMODEL;
  float* Ab = Vb + (size_t)S_LEN * DMODEL;        // [2048][1024]

  dim3 gGemm(DMODEL / 64, S_LEN / 64);            // (16, 32)
  gemm_bias_wmma<<<gGemm, 128, 0, stream>>>(x, Wq, bq, Qb, S_LEN, DMODEL, DMODEL, 1);
  gemm_bias_wmma<<<gGemm, 128, 0, stream>>>(x, Wk, bk, Kb, S_LEN, DMODEL, DMODEL, 1);
  gemm_bias_wmma<<<gGemm, 128, 0, stream>>>(x, Wv, bv, Vb, S_LEN, DMODEL, DMODEL, 1);
  spline_flash_attn<<<dim3(S_LEN / 64, HEADS), 128, 0, stream>>>(Qb, Kb, Vb, cf, Ab);
  gemm_bias_wmma<<<gGemm, 128, 0, stream>>>(Ab, Wo, bo, (float*)d_out, S_LEN, DMODEL, DMODEL, 0);
}